// Gemma4Router_20641612824865
// MI455X (gfx1250) — compile-verified
//
#include <hip/hip_runtime.h>
#include <hip/hip_bf16.h>
#include <stdint.h>

// Problem sizes (from reference)
#define HIDDEN   2048
#define NEXP     64
#define TOPK     8
#define TOKENS   16384        // 4 * 4096
#define EPSF     1e-6f
#define INV_SQRT_H 0.022097086912079608f   // 2048^-0.5

typedef float v2f __attribute__((ext_vector_type(2)));
typedef float v8f __attribute__((ext_vector_type(8)));

// ---------------------------------------------------------------------------
// Kernel 0: zero the num_tokens_per_expert region (harness poisons d_out)
// ---------------------------------------------------------------------------
__global__ void zero_counts_kernel(float* __restrict__ counts) {
    if (threadIdx.x < NEXP) counts[threadIdx.x] = 0.0f;
}

// ---------------------------------------------------------------------------
// Kernel 1: fold elementwise scale into router weights: w'[e][h] = W[e][h]*s[h]
// ---------------------------------------------------------------------------
__global__ void prep_w_kernel(const float* __restrict__ proj_w,
                              const float* __restrict__ scale,
                              float* __restrict__ wprime) {
    int i = blockIdx.x * blockDim.x + threadIdx.x;
    if (i < NEXP * HIDDEN) {
        wprime[i] = proj_w[i] * scale[i & (HIDDEN - 1)];
    }
}

// ---------------------------------------------------------------------------
// Kernel 2: per-token row factor r_t = rsqrt(mean(h^2)+eps) * H^-0.5
// One wave32 per token; 8 tokens per 256-thread block.
// ---------------------------------------------------------------------------
__global__ void __launch_bounds__(256) rms_kernel(const float* __restrict__ hs,
                                                  float* __restrict__ rms) {
    const int wave = threadIdx.x >> 5;
    const int lane = threadIdx.x & 31;
    const int t = blockIdx.x * 8 + wave;
    const float* row = hs + (size_t)t * HIDDEN;
    float s = 0.0f;
    #pragma unroll 8
    for (int k = lane; k < HIDDEN; k += 32) {
        float v = row[k];
        s += v * v;
    }
    #pragma unroll
    for (int off = 16; off > 0; off >>= 1) s += __shfl_xor(s, off, 32);
    if (lane == 0) {
        float mean = s * (1.0f / (float)HIDDEN);
        rms[t] = rsqrtf(mean + EPSF) * INV_SQRT_H;
    }
}

// ---------------------------------------------------------------------------
// Kernel 3: GEMM (WMMA f32 16x16x4) + softmax-top8 + histogram.
// Block = 16 tokens x 64 experts. 8 waves: wave w -> expert tile et=w&3,
// K-half kh=w>>2. Each wave runs 256 v_wmma_f32_16x16x4_f32 accumulations.
//
// f32 A (16x4) fragment layout (ISA 7.12.2): lane L holds row M=L%16;
// VGPR0/VGPR1 = K = kb, kb+1 with kb = (L<16)?0:2.  B (4x16): lane L holds
// col N=L%16; VGPR0/VGPR1 = K = kb, kb+1, same kb.  C/D (16x16): VGPR r at
// lane L = element (M = r + (L<16?0:8), N = L%16).
// ---------------------------------------------------------------------------
__global__ void __launch_bounds__(256) router_kernel(const float* __restrict__ hs,
                                                     const float* __restrict__ wprime,
                                                     const float* __restrict__ rms,
                                                     const float* __restrict__ pes,
                                                     float* __restrict__ out_w,
                                                     float* __restrict__ out_idx,
                                                     float* __restrict__ out_cnt) {
    __shared__ float lds_logits[16][NEXP];   // 4 KB

    const int tid  = threadIdx.x;
    const int wave = tid >> 5;
    const int lane = tid & 31;
    const int et   = wave & 3;      // expert tile (N)  0..3
    const int kh   = wave >> 2;     // K half           0..1
    const int tok0 = blockIdx.x * 16;

    // zero logits accumulator in LDS
    for (int i = tid; i < 16 * NEXP; i += 256) ((float*)lds_logits)[i] = 0.0f;
    __syncthreads();

    const int mrow = lane & 15;                 // token row (A) / expert col (B)
    const int kb   = (lane >> 4) << 1;          // 0 or 2 within each K-quad
    const float* __restrict__ arow = hs     + (size_t)(tok0 + mrow) * HIDDEN;
    const float* __restrict__ brow = wprime + (size_t)(et * 16 + mrow) * HIDDEN;

    v8f c = {};
    const int k0 = kh * (HIDDEN / 2);
    const int k1 = k0 + (HIDDEN / 2);
    #pragma unroll 4
    for (int kk = k0; kk < k1; kk += 4) {
        v2f a, b;
        a.x = arow[kk + kb];
        a.y = arow[kk + kb + 1];
        b.x = brow[kk + kb];
        b.y = brow[kk + kb + 1];
        // D = A(16x4) * B(4x16) + C, all f32
        c = __builtin_amdgcn_wmma_f32_16x16x4_f32(
                /*neg_a=*/false, a, /*neg_b=*/false, b,
                /*c_mod=*/(short)0, c, /*reuse_a=*/false, /*reuse_b=*/false);
    }

    // accumulate the two K-halves into LDS (ds_add_f32)
    const int nn    = et * 16 + (lane & 15);
    const int mbase = (lane >> 4) * 8;
    #pragma unroll
    for (int r = 0; r < 8; ++r) {
        atomicAdd(&lds_logits[mbase + r][nn], c[r]);
    }
    __syncthreads();

    // Per-token: scale by r_t, top-8 on logits (== top-8 on softmax probs),
    // weights = softmax over the selected 8, then * per_expert_scale.
    if (tid < 16) {
        const int t = tok0 + tid;
        const float f = rms[t];

        float m = -3.4e38f;
        for (int e = 0; e < NEXP; ++e) {
            float v = f * lds_logits[tid][e];
            lds_logits[tid][e] = v;            // row is private to this thread
            m = fmaxf(m, v);
        }

        uint64_t used = 0ull;
        int   idx[TOPK];
        float ex[TOPK];
        float sumexp = 0.0f;
        for (int j = 0; j < TOPK; ++j) {
            float best = -3.4e38f;
            int   bi   = 0;
            for (int e = 0; e < NEXP; ++e) {
                float v = lds_logits[tid][e];
                bool ok = !((used >> e) & 1ull) && (v > best);
                best = ok ? v : best;
                bi   = ok ? e : bi;
            }
            used  |= (1ull << bi);
            idx[j] = bi;
            ex[j]  = __expf(best - m);
            sumexp += ex[j];
        }

        const float inv = 1.0f / sumexp;
        #pragma unroll
        for (int j = 0; j < TOPK; ++j) {
            const int e = idx[j];
            out_w  [(size_t)t * TOPK + j] = ex[j] * inv * pes[e];
            out_idx[(size_t)t * TOPK + j] = (float)e;   // int32 indices as float values
            atomicAdd(&out_cnt[e], 1.0f);
        }
    }
}

// ---------------------------------------------------------------------------
// Launch
// ---------------------------------------------------------------------------
extern "C" void kernel_launch(void* const* d_in, const int* in_sizes, int n_in,
                              void* d_out, int out_size, void* d_ws, size_t ws_size,
                              hipStream_t stream) {
    (void)in_sizes; (void)n_in; (void)out_size; (void)ws_size;

    const float* hs     = (const float*)d_in[0];   // [4,4096,2048]
    const float* scale  = (const float*)d_in[1];   // [2048]
    const float* proj_w = (const float*)d_in[2];   // [64,2048]
    const float* pes    = (const float*)d_in[3];   // [64]

    float* out_w   = (float*)d_out;                 // [16384,8]
    float* out_idx = out_w + (size_t)TOKENS * TOPK; // [16384,8] (indices as floats)
    float* out_cnt = out_idx + (size_t)TOKENS * TOPK; // [64]

    float* wprime = (float*)d_ws;                   // 64*2048 floats = 512 KB
    float* rmsbuf = wprime + (size_t)NEXP * HIDDEN; // 16384 floats = 64 KB

    zero_counts_kernel<<<1, 64, 0, stream>>>(out_cnt);
    prep_w_kernel<<<(NEXP * HIDDEN) / 256, 256, 0, stream>>>(proj_w, scale, wprime);
    rms_kernel<<<TOKENS / 8, 256, 0, stream>>>(hs, rmsbuf);
    router_kernel<<<TOKENS / 16, 256, 0, stream>>>(hs, wprime, rmsbuf, pes,
                                                   out_w, out_idx, out_cnt);
}